// ClusterLookup_9311489098062
// MI455X (gfx1250) — compile-verified
//
#include <hip/hip_runtime.h>
#include <math.h>

// ---------------------------------------------------------------------------
// ClusterLookup (cosine-sim soft assignment) for MI455X / gfx1250.
// Memory-bound problem (216 MB traffic, 2.8 GFLOP => ~9.3us floor at 23.3TB/s),
// so we keep full fp32 precision and use V_WMMA_F32_16X16X4_F32 for the
// channel contraction, streaming x exactly once.
// ---------------------------------------------------------------------------

typedef float v2f __attribute__((ext_vector_type(2)));
typedef float v8f __attribute__((ext_vector_type(8)));

#define C_DIM 512
#define N_CL 27
#define N_PAD 32
#define HW 3136          // 56*56
#define TILE 16          // pixels per wave-tile
#define TILES_PER_B 196  // 3136 / 16 (exact -> tiles never cross batch)
#define WAVES_PER_BLOCK 8
#define NBLOCKS 784      // 32*196 / 8
#define NPIX 100352      // 32*56*56
#define EPS_F 1e-12f
#define ALPHA_F 2.0f

// --- Kernel 1: normalize codebook into zero-padded [32][512] workspace -----
__global__ __launch_bounds__(256) void norm_clusters_k(
    const float* __restrict__ cl, float* __restrict__ wsC) {
  __shared__ float red[256];
  const int n = blockIdx.x;     // 0..31
  const int t = threadIdx.x;    // 0..255
  if (n >= N_CL) {              // pad rows 27..31 with zeros
    wsC[n * C_DIM + t] = 0.0f;
    wsC[n * C_DIM + t + 256] = 0.0f;
    return;
  }
  const float a = cl[n * C_DIM + t];
  const float b = cl[n * C_DIM + t + 256];
  red[t] = a * a + b * b;
  __syncthreads();
  for (int s = 128; s > 0; s >>= 1) {
    if (t < s) red[t] += red[t + s];
    __syncthreads();
  }
  const float rn = 1.0f / fmaxf(sqrtf(red[0]), EPS_F);
  wsC[n * C_DIM + t] = a * rn;
  wsC[n * C_DIM + t + 256] = b * rn;
}

// --- Kernel 2: main WMMA + softmax kernel ----------------------------------
__global__ __launch_bounds__(256) void cluster_main_k(
    const float* __restrict__ x, const float* __restrict__ wsC,
    float* __restrict__ out, float* __restrict__ partials) {
  __shared__ float sh_dot[WAVES_PER_BLOCK][TILE * 33];  // stride 33: no bank conflicts
  __shared__ float sh_rn[WAVES_PER_BLOCK][TILE];
  __shared__ float sh_mx[WAVES_PER_BLOCK][TILE];
  __shared__ float sh_rz[WAVES_PER_BLOCK][TILE];
  __shared__ float sh_loss[WAVES_PER_BLOCK * TILE];

  const int wave = threadIdx.x >> 5;
  const int lid  = threadIdx.x & 31;
  const int tile = blockIdx.x * WAVES_PER_BLOCK + wave;
  const int b    = tile / TILES_PER_B;
  const int hw0  = (tile % TILES_PER_B) * TILE;
  const int pix  = lid & 15;      // A-matrix M index (pixel in tile); also B N index
  const int half = lid >> 4;      // lane half: selects K=0,1 vs K=2,3 of each quad
  const int cofs = half * 2;

  // x base: pixel 'pix' of this tile; channels strided by HW
  const float* xb  = x + (size_t)b * C_DIM * HW + hw0 + pix;
  const float* cb0 = wsC + (size_t)pix * C_DIM;         // clusters  0..15
  const float* cb1 = wsC + (size_t)(16 + pix) * C_DIM;  // clusters 16..31 (27..31 zero)

  v8f acc0 = {};
  v8f acc1 = {};
  float nsq = 0.0f;

  // K loop: 512 channels in quads of 4 -> 128 iterations x 2 WMMA
  // A 16x4 layout: lanes 0-15 hold (K=0,K=1) in (v0,v1), lanes 16-31 hold (K=2,K=3)
  // B 4x16 layout: N = lane&15 across lanes; K split by (VGPR, lane-half) same way
#pragma unroll 4
  for (int k = 0; k < C_DIM; k += 4) {
    const int c = k + cofs;
    v2f a;
    a.x = xb[(size_t)c * HW];
    a.y = xb[(size_t)(c + 1) * HW];
    nsq += a.x * a.x + a.y * a.y;             // free ||x||^2 accumulation
    const float2 t0 = *(const float2*)(cb0 + c);
    const float2 t1 = *(const float2*)(cb1 + c);
    v2f bf0; bf0.x = t0.x; bf0.y = t0.y;
    v2f bf1; bf1.x = t1.x; bf1.y = t1.y;
    acc0 = __builtin_amdgcn_wmma_f32_16x16x4_f32(false, a, false, bf0,
                                                 (short)0, acc0, false, false);
    acc1 = __builtin_amdgcn_wmma_f32_16x16x4_f32(false, a, false, bf1,
                                                 (short)0, acc1, false, false);
  }

  // combine the two half-wave partial norms (wave32 shuffle)
  nsq += __shfl_xor(nsq, 16, 32);
  if (half == 0) sh_rn[wave][pix] = 1.0f / fmaxf(sqrtf(nsq), EPS_F);

  // D layout: VGPR r holds M = r + 8*half (pixel), N = lane&15 (+16 for acc1)
#pragma unroll
  for (int r = 0; r < 8; ++r) {
    const int p = r + 8 * half;
    sh_dot[wave][p * 33 + pix]      = acc0[r];
    sh_dot[wave][p * 33 + 16 + pix] = acc1[r];
  }
  __syncthreads();

  // per-pixel softmax stats (lanes 0-15, one pixel each)
  if (half == 0) {
    const float rn = sh_rn[wave][pix];
    float v[N_CL];
    float m = -1e30f;
#pragma unroll
    for (int n = 0; n < N_CL; ++n) {
      v[n] = sh_dot[wave][pix * 33 + n] * rn;
      m = fmaxf(m, v[n]);
    }
    float Z = 0.0f, pv = 0.0f;
#pragma unroll
    for (int n = 0; n < N_CL; ++n) {
      const float e = __expf(ALPHA_F * (v[n] - m));
      Z += e;
      pv += e * v[n];
    }
    sh_mx[wave][pix] = ALPHA_F * m;
    sh_rz[wave][pix] = 1.0f / Z;
    sh_loss[wave * TILE + pix] = pv / Z;   // sum_n probs*inner for this pixel
  }
  __syncthreads();

  // write probs coalesced: for fixed n, lanes cover consecutive w (64B segments)
  float* obase = out + 1 + (size_t)b * N_CL * HW + hw0;
  for (int idx = lid; idx < TILE * N_CL; idx += 32) {
    const int p = idx & 15;
    const int n = idx >> 4;
    const float v = sh_dot[wave][p * 33 + n] * sh_rn[wave][p];
    const float prob = __expf(ALPHA_F * v - sh_mx[wave][p]) * sh_rz[wave][p];
    obase[(size_t)n * HW + p] = prob;
  }

  // deterministic per-block loss partial (fixed summation order)
  if (threadIdx.x == 0) {
    float s = 0.0f;
    for (int i = 0; i < WAVES_PER_BLOCK * TILE; ++i) s += sh_loss[i];
    partials[blockIdx.x] = s;
  }
}

// --- Kernel 3: deterministic final loss reduction --------------------------
__global__ __launch_bounds__(256) void finalize_loss_k(
    const float* __restrict__ partials, float* __restrict__ out) {
  __shared__ float red[256];
  const int t = threadIdx.x;
  float s = 0.0f;
  for (int i = t; i < NBLOCKS; i += 256) s += partials[i];
  red[t] = s;
  __syncthreads();
  if (t == 0) {
    float tot = 0.0f;
    for (int i = 0; i < 256; ++i) tot += red[i];
    out[0] = -tot / (float)NPIX;
  }
}

extern "C" void kernel_launch(void* const* d_in, const int* in_sizes, int n_in,
                              void* d_out, int out_size, void* d_ws, size_t ws_size,
                              hipStream_t stream) {
  const float* x        = (const float*)d_in[0];  // [32,512,56,56]
  const float* clusters = (const float*)d_in[1];  // [27,512]
  float* out = (float*)d_out;                     // [1 + 32*27*56*56]
  float* wsC      = (float*)d_ws;                 // 32*512 normalized clusters
  float* partials = wsC + N_PAD * C_DIM;          // 784 block partials

  norm_clusters_k<<<N_PAD, 256, 0, stream>>>(clusters, wsC);
  cluster_main_k<<<NBLOCKS, 256, 0, stream>>>(x, wsC, out, partials);
  finalize_loss_k<<<1, 256, 0, stream>>>(partials, out);
}